// DiagonalKernelAverageV2_25031069401676
// MI455X (gfx1250) — compile-verified
//
#include <hip/hip_runtime.h>
#include <cstdint>
#include <cstddef>

// ---------------------------------------------------------------------------
// DiagonalKernelAverageV2 for MI455X (gfx1250)
//
// Input : x [B=32, C=8, H=512, W=512] f32  (268 MB)
// Output: out [B, 512, 32] f32            (2 MB)
//
// Memory-bound: ~0.02 flop/byte -> HBM roofline ~11.5us @ 23.3 TB/s.
// Uses CDNA5 async global->LDS DMA (ASYNCcnt) with NT temporal hint
// (input is read-once and larger than the 192 MB L2); wave32 shuffle
// butterflies for row reductions. Two-stage (2048 WG + 256 WG) for
// occupancy / in-flight bytes; deterministic (no atomics).
// ---------------------------------------------------------------------------

#define H 512
#define W 512
#define NPLANES 256                     // B*C
#define THREADS 256
#define CHUNK_ROWS 8
#define CHUNK_FLOATS (CHUNK_ROWS * W)   // 4096 floats
#define CHUNK_BYTES  (CHUNK_FLOATS * 4) // 16 KB
#define LOADS_PER_THREAD (CHUNK_FLOATS / 4 / THREADS) // 4 x b128 per thread
#define SLABS 8
#define SLAB_ROWS (H / SLABS)           // 64

// -------------------------- CDNA5 async copy helpers -----------------------

__device__ __forceinline__ void async_load_b128(unsigned lds_addr,
                                                unsigned goffset,
                                                unsigned long long gbase) {
  // GVS mode: mem = SGPR64 base + VGPR32 offset; VDST = LDS byte address.
  // NT hint: streaming read-once data, do not retain in cache hierarchy.
  asm volatile("global_load_async_to_lds_b128 %0, %1, %2 th:TH_LOAD_NT"
               :: "v"(lds_addr), "v"(goffset), "s"(gbase) : "memory");
}

__device__ __forceinline__ unsigned lds_byte_addr(const void* p) {
  // Flat LDS pointer: low 32 bits are the LDS byte offset (ISA aperture rule).
  return (unsigned)(uintptr_t)p;
}

#define WAIT_ASYNC(n) asm volatile("s_wait_asynccnt %0" :: "n"(n) : "memory")

// -------------------- stage 1: per-slab streaming reductions ---------------
// grid = NPLANES * SLABS, block = 256.
// Writes per-row results (complete) and per-slab column partials to scratch.
// rows layout: rows[p][4][512]  : 0=rt 1=ra 2=rb 3=dm
// cols layout: cols[p][slab][3][512] : 0=ct 1=ca 2=cb

__global__ __launch_bounds__(THREADS) void
slab_kernel(const float* __restrict__ x, float* __restrict__ rows,
            float* __restrict__ cols) {
  const int wg   = blockIdx.x;
  const int p    = wg >> 3;          // plane
  const int slab = wg & 7;
  const int t    = threadIdx.x;
  const int lane = t & 31;
  const int wv   = t >> 5;           // 8 waves, one chunk-row each

  __shared__ float tile[2][CHUNK_FLOATS];   // 32 KB double buffer

  const float* slab_base =
      x + (size_t)p * (H * (size_t)W) + (size_t)slab * SLAB_ROWS * W;
  const unsigned long long gbase = (unsigned long long)(uintptr_t)slab_base;

  const int j0 = 2 * t;                     // owned columns j0, j0+1
  float ctx = 0.f, cty = 0.f, cax = 0.f, cay = 0.f, cbx = 0.f, cby = 0.f;

  // Prefetch chunk 0
  {
    const unsigned lb = lds_byte_addr(&tile[0][0]);
#pragma unroll
    for (int q = 0; q < LOADS_PER_THREAD; ++q) {
      const unsigned f16 = (unsigned)(t + THREADS * q) * 16u;
      async_load_b128(lb + f16, f16, gbase);
    }
  }

  const int NCH = SLAB_ROWS / CHUNK_ROWS;   // 8
  for (int ch = 0; ch < NCH; ++ch) {
    if (ch + 1 < NCH) {
      const unsigned lb = lds_byte_addr(&tile[(ch + 1) & 1][0]);
      const unsigned gb = (unsigned)(ch + 1) * CHUNK_BYTES;
#pragma unroll
      for (int q = 0; q < LOADS_PER_THREAD; ++q) {
        const unsigned f16 = (unsigned)(t + THREADS * q) * 16u;
        async_load_b128(lb + f16, gb + f16, gbase);
      }
      WAIT_ASYNC(LOADS_PER_THREAD);   // previous chunk's loads are done
    } else {
      WAIT_ASYNC(0);
    }
    __syncthreads();

    const float* tb = &tile[ch & 1][0];
    const int i0 = slab * SLAB_ROWS + ch * CHUNK_ROWS;  // global row base

    // ---- column-phase: accumulate ct/ca/cb for cols j0, j0+1 ----
#pragma unroll
    for (int r = 0; r < CHUNK_ROWS; ++r) {
      const int i = i0 + r;
      const float vx = tb[r * W + j0];
      const float vy = tb[r * W + j0 + 1];
      ctx += vx;                              cty += vy;
      cax += (i <= j0)            ? vx : 0.f; cay += (i <= j0 + 1)       ? vy : 0.f;
      cbx += (i <= (W - 1) - j0)  ? vx : 0.f; cby += (i <= (W - 2) - j0) ? vy : 0.f;
    }

    // ---- row-phase: wave wv reduces global row i0+wv ----
    {
      const int i = i0 + wv;
      float st = 0.f, sa = 0.f, sb = 0.f;
#pragma unroll
      for (int m = 0; m < W / 32; ++m) {
        const int col = lane + 32 * m;
        const float v = tb[wv * W + col];
        st += v;
        sa += (col <= i) ? v : 0.f;
        sb += (col <= (W - 1) - i) ? v : 0.f;
      }
#pragma unroll
      for (int off = 16; off > 0; off >>= 1) {
        st += __shfl_xor(st, off, 32);
        sa += __shfl_xor(sa, off, 32);
        sb += __shfl_xor(sb, off, 32);
      }
      if (lane == 0) {
        float* rp = rows + (size_t)p * 4 * 512;
        rp[0 * 512 + i] = st;
        rp[1 * 512 + i] = sa;
        rp[2 * 512 + i] = sb;
        rp[3 * 512 + i] = tb[wv * W + i];     // dm[i] = x[i][i]
      }
    }
    __syncthreads();
  }

  float* cp = cols + (size_t)(p * SLABS + slab) * 3 * 512;
  cp[0 * 512 + j0] = ctx;  cp[0 * 512 + j0 + 1] = cty;
  cp[1 * 512 + j0] = cax;  cp[1 * 512 + j0 + 1] = cay;
  cp[2 * 512 + j0] = cbx;  cp[2 * 512 + j0 + 1] = cby;
}

// ------------------- stage 2: combine partials, emit output ----------------
// grid = NPLANES, block = 256.

__global__ __launch_bounds__(THREADS) void
combine_kernel(const float* __restrict__ rows, const float* __restrict__ cols,
               float* __restrict__ out) {
  const int p = blockIdx.x;
  const int b = p >> 3, c = p & 7;
  const int t = threadIdx.x;

  __shared__ float ct_s[W], ca_s[W], cb_s[W];

  const int j0 = 2 * t;
  float s0 = 0.f, s1 = 0.f, s2 = 0.f, s3 = 0.f, s4 = 0.f, s5 = 0.f;
#pragma unroll
  for (int sl = 0; sl < SLABS; ++sl) {
    const float* cp = cols + (size_t)(p * SLABS + sl) * 3 * 512;
    s0 += cp[0 * 512 + j0];  s1 += cp[0 * 512 + j0 + 1];
    s2 += cp[1 * 512 + j0];  s3 += cp[1 * 512 + j0 + 1];
    s4 += cp[2 * 512 + j0];  s5 += cp[2 * 512 + j0 + 1];
  }
  ct_s[j0] = s0; ct_s[j0 + 1] = s1;
  ca_s[j0] = s2; ca_s[j0 + 1] = s3;
  cb_s[j0] = s4; cb_s[j0 + 1] = s5;
  __syncthreads();

  const float* rp = rows + (size_t)p * 4 * 512;
  const float* rt_g = rp;
  const float* ra_g = rp + 512;
  const float* rb_g = rp + 1024;
  const float* dm_g = rp + 1536;

  float* outp = out + (size_t)b * H * 32 + c;   // out[b][k][corner*8 + c]
#pragma unroll
  for (int s = 0; s < 2; ++s) {
    const int k  = t + s * THREADS;
    const int kr = (W - 1) - k;
    const float inv = 1.0f / (float)(2 * k + 1);
    const float tl = (ra_g[k] + ca_s[k] - dm_g[k]) * inv;
    const float tr = (rt_g[k] - rb_g[k] + cb_s[kr]) * inv;
    const float bl = (rb_g[kr] + ct_s[k] - cb_s[k]) * inv;
    const float br = (rt_g[kr] - ra_g[kr] + ct_s[kr] - ca_s[kr] + dm_g[kr]) * inv;
    float* o = outp + (size_t)k * 32;
    o[0]  = tl;
    o[8]  = tr;
    o[16] = bl;
    o[24] = br;
  }
}

// ------------- fallback: self-contained one-workgroup-per-plane ------------

__global__ __launch_bounds__(THREADS) void
plane_kernel(const float* __restrict__ x, float* __restrict__ out) {
  const int p = blockIdx.x;
  const int b = p >> 3, c = p & 7;
  const int t = threadIdx.x;
  const int lane = t & 31;
  const int wv = t >> 5;

  __shared__ float tile[2][CHUNK_FLOATS];                 // 32 KB
  __shared__ float rt_s[H], ra_s[H], rb_s[H], dm_s[H];    // 8 KB
  __shared__ float ct_s[W], ca_s[W], cb_s[W];             // 6 KB

  const float* plane = x + (size_t)p * (H * (size_t)W);
  const unsigned long long gbase = (unsigned long long)(uintptr_t)plane;

  const int j0 = 2 * t;
  float ctx = 0.f, cty = 0.f, cax = 0.f, cay = 0.f, cbx = 0.f, cby = 0.f;

  {
    const unsigned lb = lds_byte_addr(&tile[0][0]);
#pragma unroll
    for (int q = 0; q < LOADS_PER_THREAD; ++q) {
      const unsigned f16 = (unsigned)(t + THREADS * q) * 16u;
      async_load_b128(lb + f16, f16, gbase);
    }
  }

  const int NCH = H / CHUNK_ROWS;   // 64
  for (int ch = 0; ch < NCH; ++ch) {
    if (ch + 1 < NCH) {
      const unsigned lb = lds_byte_addr(&tile[(ch + 1) & 1][0]);
      const unsigned gb = (unsigned)(ch + 1) * CHUNK_BYTES;
#pragma unroll
      for (int q = 0; q < LOADS_PER_THREAD; ++q) {
        const unsigned f16 = (unsigned)(t + THREADS * q) * 16u;
        async_load_b128(lb + f16, gb + f16, gbase);
      }
      WAIT_ASYNC(LOADS_PER_THREAD);
    } else {
      WAIT_ASYNC(0);
    }
    __syncthreads();

    const float* tb = &tile[ch & 1][0];
    const int i0 = ch * CHUNK_ROWS;

#pragma unroll
    for (int r = 0; r < CHUNK_ROWS; ++r) {
      const int i = i0 + r;
      const float vx = tb[r * W + j0];
      const float vy = tb[r * W + j0 + 1];
      ctx += vx;                              cty += vy;
      cax += (i <= j0)            ? vx : 0.f; cay += (i <= j0 + 1)       ? vy : 0.f;
      cbx += (i <= (W - 1) - j0)  ? vx : 0.f; cby += (i <= (W - 2) - j0) ? vy : 0.f;
    }

    {
      const int i = i0 + wv;
      float st = 0.f, sa = 0.f, sb = 0.f;
#pragma unroll
      for (int m = 0; m < W / 32; ++m) {
        const int col = lane + 32 * m;
        const float v = tb[wv * W + col];
        st += v;
        sa += (col <= i) ? v : 0.f;
        sb += (col <= (W - 1) - i) ? v : 0.f;
      }
#pragma unroll
      for (int off = 16; off > 0; off >>= 1) {
        st += __shfl_xor(st, off, 32);
        sa += __shfl_xor(sa, off, 32);
        sb += __shfl_xor(sb, off, 32);
      }
      if (lane == 0) {
        rt_s[i] = st; ra_s[i] = sa; rb_s[i] = sb;
        dm_s[i] = tb[wv * W + i];
      }
    }
    __syncthreads();
  }

  ct_s[j0] = ctx; ct_s[j0 + 1] = cty;
  ca_s[j0] = cax; ca_s[j0 + 1] = cay;
  cb_s[j0] = cbx; cb_s[j0 + 1] = cby;
  __syncthreads();

  float* outp = out + (size_t)b * H * 32 + c;
#pragma unroll
  for (int s = 0; s < 2; ++s) {
    const int k  = t + s * THREADS;
    const int kr = (W - 1) - k;
    const float inv = 1.0f / (float)(2 * k + 1);
    const float tl = (ra_s[k] + ca_s[k] - dm_s[k]) * inv;
    const float tr = (rt_s[k] - rb_s[k] + cb_s[kr]) * inv;
    const float bl = (rb_s[kr] + ct_s[k] - cb_s[k]) * inv;
    const float br = (rt_s[kr] - ra_s[kr] + ct_s[kr] - ca_s[kr] + dm_s[kr]) * inv;
    float* o = outp + (size_t)k * 32;
    o[0]  = tl;
    o[8]  = tr;
    o[16] = bl;
    o[24] = br;
  }
}

// --------------------------------- launch ----------------------------------

extern "C" void kernel_launch(void* const* d_in, const int* in_sizes, int n_in,
                              void* d_out, int out_size, void* d_ws, size_t ws_size,
                              hipStream_t stream) {
  (void)in_sizes; (void)n_in; (void)out_size;
  const float* x = (const float*)d_in[0];
  float* out = (float*)d_out;

  const size_t rows_bytes = (size_t)NPLANES * 4 * 512 * sizeof(float);         // 2 MB
  const size_t cols_bytes = (size_t)NPLANES * SLABS * 3 * 512 * sizeof(float); // 12 MB

  if (d_ws != nullptr && ws_size >= rows_bytes + cols_bytes) {
    float* rows = (float*)d_ws;
    float* cols = (float*)((char*)d_ws + rows_bytes);
    slab_kernel<<<NPLANES * SLABS, THREADS, 0, stream>>>(x, rows, cols);
    combine_kernel<<<NPLANES, THREADS, 0, stream>>>(rows, cols, out);
  } else {
    plane_kernel<<<NPLANES, THREADS, 0, stream>>>(x, out);
  }
}